// GP_adapter_77704548319468
// MI455X (gfx1250) — compile-verified
//
#include <hip/hip_runtime.h>
#include <hip/hip_bf16.h>
#include <math.h>
#include <stdint.h>

// ---------------------------------------------------------------------------
// GP adapter pipeline for MI455X (gfx1250, wave32).
// Dominant cost: 4096x4096 FP32 Cholesky (~2.3e10 FLOPs). The 64MB matrix is
// L2-resident (192MB L2), so this is fp32-matrix-throughput bound, not HBM
// bound -> trailing SYRK/GEMM updates run on V_WMMA_F32_16X16X4_F32, with
// tiles staged global->LDS via GLOBAL_LOAD_ASYNC_TO_LDS_B128 (ASYNCcnt path,
// inline asm per ISA 15.18.3 since the builtin's LangAS pointer types are not
// spellable from HIP). C tiles are preloaded into the WMMA accumulators
// (B fragments negated in VALU since F32 WMMA has no A/B NEG), so writeback
// is pure stores with no RMW latency chain.
// Workspace requirement: ~68.3 MB.
// ---------------------------------------------------------------------------

#define NOBS 4096
#define NB   64          // Cholesky block size
#define NPIV 50
#define NMC  32
#define PI_F 3.14159265358979323846f
#define TSTR 68          // LDS tile row stride: 272B rows -> 16B aligned, and
                         // bank = (4*row + k) % 64 -> conflict-free frag reads

typedef float v2f __attribute__((ext_vector_type(2)));
typedef float v8f __attribute__((ext_vector_type(8)));

// 16B global -> LDS copy. Async (no VGPR round trip, ASYNCcnt) on gfx1250.
// LDS operand is the per-lane LDS byte address = low 32 bits of the generic
// pointer (LDS aperture: generic = {shared_base_hi, lds_offset[31:0]}).
static __device__ __forceinline__ void cp16(const float* g, float* l) {
#if defined(__gfx1250__)
  const uint32_t loff = (uint32_t)(uintptr_t)l;
  asm volatile("global_load_async_to_lds_b128 %0, %1, off"
               :
               : "v"(loff), "v"(g)
               : "memory");
#else
  *(float4*)l = *(const float4*)g;   // host-pass / non-gfx1250 placeholder
#endif
}

static __device__ __forceinline__ void async_fence() {
#if defined(__gfx1250__)
  asm volatile("s_wait_asynccnt 0x0" ::: "memory");
#endif
}

static __device__ __forceinline__ v8f wmma4(v2f a, v2f b, v8f c) {
  // D = A(16x4 f32) x B(4x16 f32) + C(16x16 f32)
  return __builtin_amdgcn_wmma_f32_16x16x4_f32(false, a, false, b,
                                               (short)0, c, false, false);
}

// Wave-level 32x32 subtile of a 64x64 tile, K=64. acc += A * (-B).
// As is MxK row-major. BNK=true: Bs holds B^T row-major (NxK, syrk panel);
// BNK=false: Bs holds B as KxN (solve update).
template <bool BNK>
static __device__ __forceinline__ void tile_mma64(const float (*As)[TSTR],
                                                  const float (*Bs)[TSTR],
                                                  v8f (&acc)[2][2]) {
  const int lane = threadIdx.x & 31;
  const int wave = threadIdx.x >> 5;
  const int wm = (wave >> 1) << 5;
  const int wn = (wave & 1) << 5;
  const int lm = lane & 15;
  const int kh = (lane >> 4) << 1;  // lanes 16-31 hold K=2,3 (ISA 7.12.2)
  for (int k0 = 0; k0 < NB; k0 += 4) {
    const int ka = k0 + kh;
    v2f a0, a1, b0, b1;
    a0.x = As[wm + lm][ka];      a0.y = As[wm + lm][ka + 1];
    a1.x = As[wm + 16 + lm][ka]; a1.y = As[wm + 16 + lm][ka + 1];
    if constexpr (BNK) {
      b0.x = -Bs[wn + lm][ka];      b0.y = -Bs[wn + lm][ka + 1];
      b1.x = -Bs[wn + 16 + lm][ka]; b1.y = -Bs[wn + 16 + lm][ka + 1];
    } else {
      b0.x = -Bs[ka][wn + lm];      b0.y = -Bs[ka + 1][wn + lm];
      b1.x = -Bs[ka][wn + 16 + lm]; b1.y = -Bs[ka + 1][wn + 16 + lm];
    }
    acc[0][0] = wmma4(a0, b0, acc[0][0]);
    acc[0][1] = wmma4(a0, b1, acc[0][1]);
    acc[1][0] = wmma4(a1, b0, acc[1][0]);
    acc[1][1] = wmma4(a1, b1, acc[1][1]);
  }
}

// Preload / store the 64x64 C tile in WMMA C/D fragment layout (ISA 7.12.2:
// VGPR r holds M = r (+8 for lanes 16-31), N = lane&15).
static __device__ __forceinline__ void preload_C(const float* C, int ldc,
                                                 int rb, int cb, v8f (&acc)[2][2]) {
  const int lane = threadIdx.x & 31, wave = threadIdx.x >> 5;
  const int wm = (wave >> 1) << 5, wn = (wave & 1) << 5;
  const int lm = lane & 15, mo = (lane >> 4) << 3;
  for (int r = 0; r < 8; ++r) {
    const size_t row0 = (size_t)(rb + wm + mo + r) * ldc;
    const size_t row1 = (size_t)(rb + wm + 16 + mo + r) * ldc;
    acc[0][0][r] = C[row0 + cb + wn + lm];
    acc[0][1][r] = C[row0 + cb + wn + 16 + lm];
    acc[1][0][r] = C[row1 + cb + wn + lm];
    acc[1][1][r] = C[row1 + cb + wn + 16 + lm];
  }
}

static __device__ __forceinline__ void store_C(float* C, int ldc,
                                               int rb, int cb, v8f (&acc)[2][2]) {
  const int lane = threadIdx.x & 31, wave = threadIdx.x >> 5;
  const int wm = (wave >> 1) << 5, wn = (wave & 1) << 5;
  const int lm = lane & 15, mo = (lane >> 4) << 3;
  for (int r = 0; r < 8; ++r) {
    const size_t row0 = (size_t)(rb + wm + mo + r) * ldc;
    const size_t row1 = (size_t)(rb + wm + 16 + mo + r) * ldc;
    C[row0 + cb + wn + lm]      = acc[0][0][r];
    C[row0 + cb + wn + 16 + lm] = acc[0][1][r];
    C[row1 + cb + wn + lm]      = acc[1][0][r];
    C[row1 + cb + wn + 16 + lm] = acc[1][1][r];
  }
}

// ---------------------------------------------------------------------------
// Ktt build: K(i,j) = exp(lv - 2*exp(lt)*sin^2(pi*(ti-tj)/P)) + diag terms
// ---------------------------------------------------------------------------
__global__ void k_build_Ktt(float* __restrict__ L, const float* __restrict__ mjd,
                            const float* __restrict__ err, const float* __restrict__ P,
                            const float* __restrict__ lv, const float* __restrict__ lt,
                            const float* __restrict__ lvl) {
  const size_t idx = (size_t)blockIdx.x * blockDim.x + threadIdx.x;
  const int i = (int)(idx >> 12);
  const int j = (int)(idx & (NOBS - 1));
  const float pinv = PI_F / P[0];
  const float tt = 2.0f * expf(lt[0]);
  const float s = sinf(pinv * (mjd[i] - mjd[j]));
  float v = expf(lv[0] - tt * s * s);
  if (i == j) v += err[i] * err[i] + expf(lvl[0]);
  L[idx] = v;
}

// RHS buffer X[4096][64]: cols 0..49 = Ktx, col 50 = mag, rest zero.
__global__ void k_build_X(float* __restrict__ X, const float* __restrict__ mjd,
                          const float* __restrict__ mag, const float* __restrict__ P,
                          const float* __restrict__ lv, const float* __restrict__ lt) {
  const int r = blockIdx.x;
  const int c = threadIdx.x;  // 64 threads
  float v = 0.0f;
  if (c < NPIV) {
    const float reg = (10.0f / 49.0f) * (float)c;
    const float s = sinf((PI_F / P[0]) * (mjd[r] - reg));
    v = expf(lv[0] - 2.0f * expf(lt[0]) * s * s);
  } else if (c == NPIV) {
    v = mag[r];
  }
  X[r * 64 + c] = v;
}

// ---------------------------------------------------------------------------
// Blocked Cholesky: potf2 (diag block), trsm (panel), WMMA syrk (trailing)
// ---------------------------------------------------------------------------
__global__ void k_potf2(float* __restrict__ L, int kb) {
  __shared__ float T[NB][NB + 1];
  for (int t = threadIdx.x; t < NB * NB; t += blockDim.x) {
    const int r = t >> 6, c = t & (NB - 1);
    T[r][c] = L[(size_t)(kb + r) * NOBS + kb + c];
  }
  __syncthreads();
  for (int j = 0; j < NB; ++j) {
    if (threadIdx.x == 0) T[j][j] = sqrtf(T[j][j]);
    __syncthreads();
    const float d = T[j][j];
    for (int i = j + 1 + (int)threadIdx.x; i < NB; i += blockDim.x) T[i][j] /= d;
    __syncthreads();
    const int rem = NB - j - 1;
    for (int t = threadIdx.x; t < rem * rem; t += blockDim.x) {
      const int i = j + 1 + t / rem, l = j + 1 + t % rem;
      if (l <= i) T[i][l] -= T[i][j] * T[l][j];
    }
    __syncthreads();
  }
  for (int t = threadIdx.x; t < NB * NB; t += blockDim.x) {
    const int r = t >> 6, c = t & (NB - 1);
    if (c <= r) L[(size_t)(kb + r) * NOBS + kb + c] = T[r][c];
  }
}

// Panel solve: A[i,k] <- A[i,k] * Lkk^-T, one 64-row tile per block, 64 thr.
__global__ void k_trsm(float* __restrict__ L, int kb) {
  __shared__ float Lk[NB][NB + 1];
  __shared__ float Bt[NB][NB + 1];
  const int rowbase = kb + NB + blockIdx.x * NB;
  for (int t = threadIdx.x; t < NB * NB; t += blockDim.x) {
    const int r = t >> 6, c = t & (NB - 1);
    Lk[r][c] = L[(size_t)(kb + r) * NOBS + kb + c];
    Bt[r][c] = L[(size_t)(rowbase + r) * NOBS + kb + c];
  }
  __syncthreads();
  const int r = threadIdx.x;  // one row per thread
  for (int j = 0; j < NB; ++j) {
    float s = Bt[r][j];
    for (int m = 0; m < j; ++m) s -= Bt[r][m] * Lk[j][m];
    Bt[r][j] = s / Lk[j][j];
  }
  __syncthreads();
  for (int t = threadIdx.x; t < NB * NB; t += blockDim.x) {
    const int rr = t >> 6, c = t & (NB - 1);
    L[(size_t)(rowbase + rr) * NOBS + kb + c] = Bt[rr][c];
  }
}

// Trailing update (SYRK): A[i,j] -= A[i,k]*A[j,k]^T over lower tiles, WMMA.
// Both panels staged row-major (MxK) via async-to-LDS; B consumed as N x K.
__global__ void k_syrk(float* __restrict__ L, int kb) {
  __shared__ float As[NB][TSTR];
  __shared__ float Bs[NB][TSTR];
  int b = blockIdx.x, ti = 0;
  while (b >= ti + 1) { b -= ti + 1; ++ti; }  // lower-triangular tile decode
  const int tj = b;
  const int trail = kb + NB;
  const int rb = trail + ti * NB, cb = trail + tj * NB;
  for (int t = threadIdx.x; t < NB * 16; t += blockDim.x) {
    const int m = t >> 4, c4 = (t & 15) << 2;   // 16B chunks, 16B aligned
    cp16(&L[(size_t)(rb + m) * NOBS + kb + c4], &As[m][c4]);
    cp16(&L[(size_t)(cb + m) * NOBS + kb + c4], &Bs[m][c4]);
  }
  v8f acc[2][2];
  preload_C(L, NOBS, rb, cb, acc);   // overlaps with async staging
  async_fence();
  __syncthreads();
  tile_mma64<true>(As, Bs, acc);     // acc = C - Ai * Aj^T
  store_C(L, NOBS, rb, cb, acc);     // pure stores, no RMW
}

// ---------------------------------------------------------------------------
// Blocked forward substitution L X = [Ktx | mag]  (X: 4096 x 64)
// ---------------------------------------------------------------------------
__global__ void k_solveX_diag(const float* __restrict__ L, float* __restrict__ X, int kb) {
  __shared__ float Lk[NB][NB + 1];
  __shared__ float Bs[NB][NB + 1];
  for (int t = threadIdx.x; t < NB * NB; t += blockDim.x) {
    const int r = t >> 6, c = t & (NB - 1);
    Lk[r][c] = L[(size_t)(kb + r) * NOBS + kb + c];
    Bs[r][c] = X[(kb + r) * 64 + c];
  }
  __syncthreads();
  const int c = threadIdx.x;  // 64 threads, one RHS column each
  for (int j = 0; j < NB; ++j) {
    const float x = Bs[j][c] / Lk[j][j];
    Bs[j][c] = x;
    for (int m = j + 1; m < NB; ++m) Bs[m][c] -= Lk[m][j] * x;
  }
  __syncthreads();
  for (int t = threadIdx.x; t < NB * NB; t += blockDim.x) {
    const int r = t >> 6, cc = t & (NB - 1);
    X[(kb + r) * 64 + cc] = Bs[r][cc];
  }
}

// GEMM update: X[i] -= L[i,k] * X[k], 64x64 tile per block, WMMA.
__global__ void k_updX(const float* __restrict__ L, float* __restrict__ X, int kb) {
  __shared__ float As[NB][TSTR];   // M x K
  __shared__ float Bs[NB][TSTR];   // K x N (X rows are contiguous)
  const int rb = kb + NB + blockIdx.x * NB;
  for (int t = threadIdx.x; t < NB * 16; t += blockDim.x) {
    const int m = t >> 4, c4 = (t & 15) << 2;
    cp16(&L[(size_t)(rb + m) * NOBS + kb + c4], &As[m][c4]);
    cp16(&X[(kb + m) * 64 + c4], &Bs[m][c4]);
  }
  v8f acc[2][2];
  preload_C(X, 64, rb, 0, acc);
  async_fence();
  __syncthreads();
  tile_mma64<false>(As, Bs, acc);    // acc = C - L[i,k] * X[k]
  store_C(X, 64, rb, 0, acc);
}

// ---------------------------------------------------------------------------
// S = Kxx - A^T A (+ jitter I), mu = A^T b   (grid 51x51, 64 threads)
// ---------------------------------------------------------------------------
__global__ void k_gram(const float* __restrict__ X, float* __restrict__ S,
                       float* __restrict__ mu, const float* __restrict__ P,
                       const float* __restrict__ lv, const float* __restrict__ lt) {
  const int a = blockIdx.x, bc = blockIdx.y;
  __shared__ float red[64];
  float s = 0.0f;
  for (int r = threadIdx.x; r < NOBS; r += 64) s += X[r * 64 + a] * X[r * 64 + bc];
  red[threadIdx.x] = s;
  __syncthreads();
  for (int off = 32; off > 0; off >>= 1) {
    if ((int)threadIdx.x < off) red[threadIdx.x] += red[threadIdx.x + off];
    __syncthreads();
  }
  if (threadIdx.x == 0) {
    if (a < NPIV && bc < NPIV) {
      const float ra = (10.0f / 49.0f) * (float)a;
      const float rb = (10.0f / 49.0f) * (float)bc;
      const float sn = sinf((PI_F / P[0]) * (ra - rb));
      const float kxx = expf(lv[0] - 2.0f * expf(lt[0]) * sn * sn);
      S[a * 64 + bc] = kxx - red[0] + ((a == bc) ? 1e-5f : 0.0f);
    } else if (bc == NPIV && a < NPIV) {
      mu[a] = red[0];
    }
  }
}

__global__ void k_chol_small(const float* __restrict__ S, float* __restrict__ Lp) {
  __shared__ float T[NPIV][NPIV + 1];
  for (int t = threadIdx.x; t < NPIV * NPIV; t += 64) T[t / NPIV][t % NPIV] = S[(t / NPIV) * 64 + t % NPIV];
  __syncthreads();
  for (int j = 0; j < NPIV; ++j) {
    if (threadIdx.x == 0) T[j][j] = sqrtf(T[j][j]);
    __syncthreads();
    const float d = T[j][j];
    for (int i = j + 1 + (int)threadIdx.x; i < NPIV; i += 64) T[i][j] /= d;
    __syncthreads();
    const int rem = NPIV - j - 1;
    for (int t = threadIdx.x; t < rem * rem; t += 64) {
      const int i = j + 1 + t / rem, l = j + 1 + t % rem;
      if (l <= i) T[i][l] -= T[i][j] * T[l][j];
    }
    __syncthreads();
  }
  for (int t = threadIdx.x; t < NPIV * 64; t += 64) {
    const int r = t >> 6, c = t & 63;
    Lp[r * 64 + c] = (c <= r && c < NPIV) ? T[r][c] : 0.0f;
  }
}

// ---------------------------------------------------------------------------
// z = mu + eps @ Lp^T ; conv1d(w=3) + relu ; maxpool(2,2,pad(1,1)) ;
// fc1 + relu ; fc2 ; log_softmax  -> out (32 x 2)
// ---------------------------------------------------------------------------
__global__ void k_head(const float* __restrict__ Lp, const float* __restrict__ mu,
                       const float* __restrict__ eps, const float* __restrict__ conv_w,
                       const float* __restrict__ conv_b, const float* __restrict__ fc1_w,
                       const float* __restrict__ fc1_b, const float* __restrict__ fc2_w,
                       const float* __restrict__ fc2_b, float* __restrict__ out) {
  __shared__ float z[NMC][NPIV];
  __shared__ float pooled[NMC][8][25];
  __shared__ float h1[NMC][64];
  for (int t = threadIdx.x; t < NMC * NPIV; t += blockDim.x) {
    const int m = t / NPIV, p = t % NPIV;
    float s = mu[p];
    for (int q = 0; q < NPIV; ++q) s += eps[m * NPIV + q] * Lp[p * 64 + q];
    z[m][p] = s;
  }
  __syncthreads();
  for (int t = threadIdx.x; t < NMC * 8 * 25; t += blockDim.x) {
    const int m = t / 200, rem = t % 200, o = rem / 25, p = rem % 25;
    float mx = -1e30f;
    for (int h = 2 * p - 1; h <= 2 * p; ++h) {
      if (h < 0 || h > 47) continue;
      float c = conv_b[o];
      for (int i = 0; i < 3; ++i) c += conv_w[o * 3 + i] * z[m][h + i];
      mx = fmaxf(mx, fmaxf(c, 0.0f));
    }
    pooled[m][o][p] = mx;
  }
  __syncthreads();
  for (int t = threadIdx.x; t < NMC * 64; t += blockDim.x) {
    const int m = t >> 6, u = t & 63;
    float s = fc1_b[u];
    const float* pm = &pooled[m][0][0];
    for (int f = 0; f < 200; ++f) s += pm[f] * fc1_w[u * 200 + f];
    h1[m][u] = fmaxf(s, 0.0f);
  }
  __syncthreads();
  for (int m = threadIdx.x; m < NMC; m += blockDim.x) {
    float l0 = fc2_b[0], l1 = fc2_b[1];
    for (int u = 0; u < 64; ++u) {
      l0 += h1[m][u] * fc2_w[u];
      l1 += h1[m][u] * fc2_w[64 + u];
    }
    const float mx = fmaxf(l0, l1);
    const float lse = mx + logf(expf(l0 - mx) + expf(l1 - mx));
    out[m * 2 + 0] = l0 - lse;
    out[m * 2 + 1] = l1 - lse;
  }
}

// ---------------------------------------------------------------------------
extern "C" void kernel_launch(void* const* d_in, const int* in_sizes, int n_in,
                              void* d_out, int out_size, void* d_ws, size_t ws_size,
                              hipStream_t stream) {
  const float* mjd   = (const float*)d_in[0];
  const float* mag   = (const float*)d_in[1];
  const float* err   = (const float*)d_in[2];
  const float* P     = (const float*)d_in[3];
  const float* eps   = (const float*)d_in[4];
  const float* lv    = (const float*)d_in[5];
  const float* lt    = (const float*)d_in[6];
  const float* lvl   = (const float*)d_in[7];
  const float* conv_w = (const float*)d_in[8];
  const float* conv_b = (const float*)d_in[9];
  const float* fc1_w = (const float*)d_in[10];
  const float* fc1_b = (const float*)d_in[11];
  const float* fc2_w = (const float*)d_in[12];
  const float* fc2_b = (const float*)d_in[13];

  // Workspace layout (floats). Requires ws_size >= ~68.3 MB.
  float* L  = (float*)d_ws;                       // 4096*4096
  float* X  = L + (size_t)NOBS * NOBS;            // 4096*64
  float* S  = X + (size_t)NOBS * 64;              // 64*64
  float* Lp = S + 64 * 64;                        // 64*64
  float* mu = Lp + 64 * 64;                       // 64

  k_build_Ktt<<<(NOBS * (NOBS / 256)), 256, 0, stream>>>(L, mjd, err, P, lv, lt, lvl);
  k_build_X<<<NOBS, 64, 0, stream>>>(X, mjd, mag, P, lv, lt);

  const int NBLK = NOBS / NB;  // 64
  for (int k = 0; k < NBLK; ++k) {
    const int kb = k * NB;
    k_potf2<<<1, 256, 0, stream>>>(L, kb);
    const int T = NBLK - 1 - k;  // remaining 64-row tiles
    if (T > 0) {
      k_trsm<<<T, NB, 0, stream>>>(L, kb);
      k_syrk<<<T * (T + 1) / 2, 128, 0, stream>>>(L, kb);
    }
  }
  for (int k = 0; k < NBLK; ++k) {
    const int kb = k * NB;
    k_solveX_diag<<<1, 64, 0, stream>>>(L, X, kb);
    const int T = NBLK - 1 - k;
    if (T > 0) k_updX<<<T, 128, 0, stream>>>(L, X, kb);
  }

  dim3 gg(NPIV + 1, NPIV + 1);
  k_gram<<<gg, 64, 0, stream>>>(X, S, mu, P, lv, lt);
  k_chol_small<<<1, 64, 0, stream>>>(S, Lp);
  k_head<<<1, 256, 0, stream>>>(Lp, mu, eps, conv_w, conv_b, fc1_w, fc1_b,
                                fc2_w, fc2_b, (float*)d_out);
}